// GATv2_layer_53807350284430
// MI455X (gfx1250) — compile-verified
//
#include <hip/hip_runtime.h>
#include <hip/hip_bf16.h>
#include <math.h>

// ---------------------------------------------------------------------------
// GATv2 2-layer GNN for MI455X (gfx1250, wave32).
// Dense node transforms use V_WMMA_F32_16X16X4_F32 (exact fp32 path).
// Edge phases are gather/scatter bound; working set (~115MB) fits in 192MB L2.
// ---------------------------------------------------------------------------

typedef __attribute__((ext_vector_type(2))) float v2f;
typedef __attribute__((ext_vector_type(8))) float v8f;

#define NEG_SLOPE 0.2f

// float atomic max via sign-split trick -> native int/uint atomics in L2
__device__ __forceinline__ void atomicMaxF32(float* addr, float val) {
  if (val >= 0.0f) {
    atomicMax((int*)addr, __float_as_int(val));
  } else {
    atomicMin((unsigned int*)addr, __float_as_uint(val));
  }
}

// row gather: b128 loads when the row is 16B-aligned-and-sized, else scalar
template <int L>
__device__ __forceinline__ void load_row(const float* __restrict__ p, float* r) {
  if constexpr ((L % 4) == 0) {
    const float4* p4 = (const float4*)p;
#pragma unroll
    for (int i = 0; i < L / 4; ++i) {
      const float4 v = p4[i];
      r[4 * i + 0] = v.x; r[4 * i + 1] = v.y;
      r[4 * i + 2] = v.z; r[4 * i + 3] = v.w;
    }
  } else {
#pragma unroll
    for (int i = 0; i < L; ++i) r[i] = p[i];
  }
}

// ---------------------------------------------------------------------------
// generic fill
// ---------------------------------------------------------------------------
__global__ void fill_f32(float* __restrict__ p, float v, int n) {
  int i = blockIdx.x * blockDim.x + threadIdx.x;
  if (i < n) p[i] = v;
}

// ---------------------------------------------------------------------------
// Layer-1 transform: xl = x @ Wl1, xr = x @ Wr1   ([N,3] x [3,32])
// One 16-row tile per wave. K=3 padded to 4 for V_WMMA_F32_16X16X4_F32.
// A layout (32-bit 16x4): lanes 0-15 -> K={0,1} in v0/v1; lanes 16-31 -> K={2,3}
// D layout: vgpr g, lane l -> row = g + 8*(l>=16), col = l&15
// ---------------------------------------------------------------------------
__global__ void gemm_xfm_l1(const float* __restrict__ x,
                            const float* __restrict__ Wl,
                            const float* __restrict__ Wr,
                            float* __restrict__ xl,
                            float* __restrict__ xr, int N) {
  const int wave = threadIdx.x >> 5;
  const int lane = threadIdx.x & 31;
  const int tile = blockIdx.x * (blockDim.x >> 5) + wave;
  const int row0 = tile * 16;
  if (row0 >= N) return;                    // wave-uniform; EXEC stays full

  const int half = lane >> 4;               // 0: K={0,1}  1: K={2,3}
  const int lr   = lane & 15;
  const int k0   = half * 2;

  int arow = row0 + lr;
  if (arow >= N) arow = N - 1;              // clamp load; tail masked at store

  v2f a;
  a.x = x[(size_t)arow * 3 + k0];                         // k0 = 0 or 2
  a.y = (k0 + 1 < 3) ? x[(size_t)arow * 3 + k0 + 1] : 0.0f;

  // B tiles (4x16 each): cols [0,16) and [16,32)
  v2f bl0, bl1, br0, br1;
  {
    const int n0 = lr, n1 = 16 + lr;
    const bool k1ok = (k0 + 1 < 3);
    bl0.x = Wl[k0 * 32 + n0]; bl0.y = k1ok ? Wl[(k0 + 1) * 32 + n0] : 0.0f;
    bl1.x = Wl[k0 * 32 + n1]; bl1.y = k1ok ? Wl[(k0 + 1) * 32 + n1] : 0.0f;
    br0.x = Wr[k0 * 32 + n0]; br0.y = k1ok ? Wr[(k0 + 1) * 32 + n0] : 0.0f;
    br1.x = Wr[k0 * 32 + n1]; br1.y = k1ok ? Wr[(k0 + 1) * 32 + n1] : 0.0f;
  }

  v8f cl0 = {}, cl1 = {}, cr0 = {}, cr1 = {};
  cl0 = __builtin_amdgcn_wmma_f32_16x16x4_f32(false, a, false, bl0, (short)0, cl0, false, false);
  cl1 = __builtin_amdgcn_wmma_f32_16x16x4_f32(false, a, false, bl1, (short)0, cl1, false, false);
  cr0 = __builtin_amdgcn_wmma_f32_16x16x4_f32(false, a, false, br0, (short)0, cr0, false, false);
  cr1 = __builtin_amdgcn_wmma_f32_16x16x4_f32(false, a, false, br1, (short)0, cr1, false, false);

  const int rbase = row0 + 8 * half;
  if (row0 + 16 <= N) {
    // fast path (all tiles except the last): unconditional store clauses
#pragma unroll
    for (int g = 0; g < 8; ++g) {
      const size_t r = (size_t)(rbase + g);
      xl[r * 32 + lr]      = cl0[g];
      xl[r * 32 + 16 + lr] = cl1[g];
      xr[r * 32 + lr]      = cr0[g];
      xr[r * 32 + 16 + lr] = cr1[g];
    }
  } else {
#pragma unroll
    for (int g = 0; g < 8; ++g) {
      const int r = rbase + g;
      if (r < N) {
        xl[(size_t)r * 32 + lr]      = cl0[g];
        xl[(size_t)r * 32 + 16 + lr] = cl1[g];
        xr[(size_t)r * 32 + lr]      = cr0[g];
        xr[(size_t)r * 32 + 16 + lr] = cr1[g];
      }
    }
  }
}

// ---------------------------------------------------------------------------
// Layer-2 transform: xl2 = h @ Wl2, xr2 = h @ Wr2  ([N,32] x [32,6], pad N->16)
// 8 accumulating K-steps of WMMA_F32_16X16X4_F32, x2 weight matrices.
// ---------------------------------------------------------------------------
__global__ void gemm_xfm_l2(const float* __restrict__ h,
                            const float* __restrict__ Wl,
                            const float* __restrict__ Wr,
                            float* __restrict__ xl,
                            float* __restrict__ xr, int N) {
  const int wave = threadIdx.x >> 5;
  const int lane = threadIdx.x & 31;
  const int tile = blockIdx.x * (blockDim.x >> 5) + wave;
  const int row0 = tile * 16;
  if (row0 >= N) return;

  const int half = lane >> 4;
  const int lr   = lane & 15;
  const int k0   = half * 2;
  const bool ncol = (lr < 6);

  int arow = row0 + lr;
  if (arow >= N) arow = N - 1;

  v8f cl = {}, cr = {};
#pragma unroll
  for (int kb = 0; kb < 8; ++kb) {
    const int kk = kb * 4 + k0;                 // K index for this lane half
    v2f a;
    a.x = h[(size_t)arow * 32 + kk];
    a.y = h[(size_t)arow * 32 + kk + 1];
    v2f bl, br;
    bl.x = ncol ? Wl[kk * 6 + lr] : 0.0f;
    bl.y = ncol ? Wl[(kk + 1) * 6 + lr] : 0.0f;
    br.x = ncol ? Wr[kk * 6 + lr] : 0.0f;
    br.y = ncol ? Wr[(kk + 1) * 6 + lr] : 0.0f;
    cl = __builtin_amdgcn_wmma_f32_16x16x4_f32(false, a, false, bl, (short)0, cl, false, false);
    cr = __builtin_amdgcn_wmma_f32_16x16x4_f32(false, a, false, br, (short)0, cr, false, false);
  }

  const int rbase = row0 + 8 * half;
  if (row0 + 16 <= N) {
    if (ncol) {                               // single divergent guard
#pragma unroll
      for (int g = 0; g < 8; ++g) {
        const size_t r = (size_t)(rbase + g);
        xl[r * 6 + lr] = cl[g];
        xr[r * 6 + lr] = cr[g];
      }
    }
  } else {
#pragma unroll
    for (int g = 0; g < 8; ++g) {
      const int r = rbase + g;
      if (r < N && ncol) {
        xl[(size_t)r * 6 + lr] = cl[g];
        xr[(size_t)r * 6 + lr] = cr[g];
      }
    }
  }
}

// ---------------------------------------------------------------------------
// Edge pass A: raw attention scores + segment max (per dst, per head)
// Virtual edge list: e < E -> edge_index, e >= E -> self loop (e-E).
// ---------------------------------------------------------------------------
template <int H, int C>
__global__ void edge_scores(const float* __restrict__ xl,
                            const float* __restrict__ xr,
                            const float* __restrict__ att,
                            const int* __restrict__ src,
                            const int* __restrict__ dst,
                            int E, int N,
                            float* __restrict__ alpha,
                            float* __restrict__ m) {
  const int e = blockIdx.x * blockDim.x + threadIdx.x;
  const int Ep = E + N;
  if (e >= Ep) return;
  int s, d;
  if (e < E) { s = src[e]; d = dst[e]; } else { s = d = e - E; }

  float xs[H * C], xd[H * C];
  load_row<H * C>(xl + (size_t)s * H * C, xs);      // b128 gathers
  load_row<H * C>(xr + (size_t)d * H * C, xd);

  float sc[H];
#pragma unroll
  for (int h = 0; h < H; ++h) {
    float acc = 0.0f;
#pragma unroll
    for (int c = 0; c < C; ++c) {
      float v = xs[h * C + c] + xd[h * C + c];
      v = (v > 0.0f) ? v : NEG_SLOPE * v;           // LeakyReLU(0.2)
      acc = fmaf(att[h * C + c], v, acc);
    }
    sc[h] = acc;
    atomicMaxF32(&m[(size_t)d * H + h], acc);
  }
  if constexpr (H == 4) {
    *(float4*)(alpha + (size_t)e * 4) = make_float4(sc[0], sc[1], sc[2], sc[3]);
  } else {
#pragma unroll
    for (int h = 0; h < H; ++h) alpha[(size_t)e * H + h] = sc[h];
  }
}

// ---------------------------------------------------------------------------
// Edge pass B: exp(score - max) + segment sum
// ---------------------------------------------------------------------------
template <int H>
__global__ void edge_exp(float* __restrict__ alpha,
                         const float* __restrict__ m,
                         float* __restrict__ ssum,
                         const int* __restrict__ dst,
                         int E, int N) {
  const int e = blockIdx.x * blockDim.x + threadIdx.x;
  const int Ep = E + N;
  if (e >= Ep) return;
  const int d = (e < E) ? dst[e] : (e - E);

  float av[H], mv[H];
  load_row<H>(alpha + (size_t)e * H, av);           // b128 when H==4
  load_row<H>(m + (size_t)d * H, mv);
#pragma unroll
  for (int h = 0; h < H; ++h) {
    av[h] = expf(av[h] - mv[h]);
    atomicAdd(&ssum[(size_t)d * H + h], av[h]);
  }
  if constexpr (H == 4) {
    *(float4*)(alpha + (size_t)e * 4) = make_float4(av[0], av[1], av[2], av[3]);
  } else {
#pragma unroll
    for (int h = 0; h < H; ++h) alpha[(size_t)e * H + h] = av[h];
  }
}

// ---------------------------------------------------------------------------
// Edge pass C: normalized weighted scatter-add of source features
// ---------------------------------------------------------------------------
template <int H, int C>
__global__ void edge_scatter(const float* __restrict__ xl,
                             const float* __restrict__ alpha,
                             const float* __restrict__ ssum,
                             const int* __restrict__ src,
                             const int* __restrict__ dst,
                             int E, int N,
                             float* __restrict__ out) {
  const int e = blockIdx.x * blockDim.x + threadIdx.x;
  const int Ep = E + N;
  if (e >= Ep) return;
  int s, d;
  if (e < E) { s = src[e]; d = dst[e]; } else { s = d = e - E; }

  float xs[H * C], av[H], sv[H];
  load_row<H * C>(xl + (size_t)s * H * C, xs);
  load_row<H>(alpha + (size_t)e * H, av);
  load_row<H>(ssum + (size_t)d * H, sv);
#pragma unroll
  for (int h = 0; h < H; ++h) {
    const float a = av[h] / sv[h];
#pragma unroll
    for (int c = 0; c < C; ++c) {
      atomicAdd(&out[(size_t)d * H * C + h * C + c], xs[h * C + c] * a);
    }
  }
}

// ---------------------------------------------------------------------------
// Layer-1 epilogue: h = ELU(acc + b1)
// ---------------------------------------------------------------------------
__global__ void bias_elu32(float* __restrict__ hbuf, const float* __restrict__ b, int N) {
  const int i = blockIdx.x * blockDim.x + threadIdx.x;
  if (i < N * 32) {
    const float v = hbuf[i] + b[i & 31];
    hbuf[i] = (v > 0.0f) ? v : expm1f(v);
  }
}

// ---------------------------------------------------------------------------
// Global mean pool (sums + counts); b2 folded into finalize
// ---------------------------------------------------------------------------
__global__ void pool_sum(const float* __restrict__ h2, const int* __restrict__ batch,
                         int N, float* __restrict__ pooled, float* __restrict__ cnt) {
  const int n = blockIdx.x * blockDim.x + threadIdx.x;
  if (n >= N) return;
  const int g = batch[n];
#pragma unroll
  for (int c = 0; c < 6; ++c)
    atomicAdd(&pooled[(size_t)g * 6 + c], h2[(size_t)n * 6 + c]);
  atomicAdd(&cnt[g], 1.0f);
}

// mean(h)+b2, then log_softmax over 6 classes
__global__ void finalize(const float* __restrict__ pooled, const float* __restrict__ cnt,
                         const float* __restrict__ b2, float* __restrict__ out, int G) {
  const int g = blockIdx.x * blockDim.x + threadIdx.x;
  if (g >= G) return;
  const float inv = 1.0f / fmaxf(cnt[g], 1.0f);
  float v[6], mx = -INFINITY;
#pragma unroll
  for (int c = 0; c < 6; ++c) {
    v[c] = pooled[(size_t)g * 6 + c] * inv + b2[c];
    mx = fmaxf(mx, v[c]);
  }
  float s = 0.0f;
#pragma unroll
  for (int c = 0; c < 6; ++c) s += expf(v[c] - mx);
  const float lse = mx + logf(s);
#pragma unroll
  for (int c = 0; c < 6; ++c) out[(size_t)g * 6 + c] = v[c] - lse;
}

// ---------------------------------------------------------------------------
// host launcher
// ---------------------------------------------------------------------------
extern "C" void kernel_launch(void* const* d_in, const int* in_sizes, int n_in,
                              void* d_out, int out_size, void* d_ws, size_t ws_size,
                              hipStream_t stream) {
  const float* x    = (const float*)d_in[0];
  const int*   ei   = (const int*)d_in[1];
  const int*   batch= (const int*)d_in[2];
  // d_in[3] = n_graphs scalar (device); G recovered from out_size
  const float* Wl1  = (const float*)d_in[4];
  const float* Wr1  = (const float*)d_in[5];
  const float* att1 = (const float*)d_in[6];
  const float* b1   = (const float*)d_in[7];
  const float* Wl2  = (const float*)d_in[8];
  const float* Wr2  = (const float*)d_in[9];
  const float* att2 = (const float*)d_in[10];
  const float* b2   = (const float*)d_in[11];

  const int N  = in_sizes[0] / 3;
  const int E  = in_sizes[1] / 2;
  const int G  = out_size / 6;
  const int Ep = E + N;
  const int* src = ei;
  const int* dst = ei + E;
  float* out = (float*)d_out;

  // workspace carve-up (256B aligned blocks; all row strides keep 16B align
  // where the b128 paths are used: 128B rows for layer-1 features, 16B alpha)
  char* ws = (char*)d_ws;
  size_t off = 0;
  auto alloc = [&](size_t bytes) -> float* {
    float* p = (float*)(ws + off);
    off += (bytes + 255) & ~(size_t)255;
    return p;
  };
  float* xl1    = alloc((size_t)N * 32 * 4);
  float* xr1    = alloc((size_t)N * 32 * 4);
  float* alpha1 = alloc((size_t)Ep * 4 * 4);
  float* m1     = alloc((size_t)N * 4 * 4);
  float* s1     = alloc((size_t)N * 4 * 4);
  float* acc1   = alloc((size_t)N * 32 * 4);   // layer-1 out, becomes h in place
  float* xl2    = alloc((size_t)N * 6 * 4);
  float* xr2    = alloc((size_t)N * 6 * 4);
  float* alpha2 = alloc((size_t)Ep * 4);
  float* m2     = alloc((size_t)N * 4);
  float* s2     = alloc((size_t)N * 4);
  float* acc2   = alloc((size_t)N * 6 * 4);
  float* pooled = alloc((size_t)G * 6 * 4);
  float* cnt    = alloc((size_t)G * 4);

  const int B = 256;
  auto grid1 = [&](long n) { return dim3((unsigned)((n + B - 1) / B)); };

  // deterministic re-init every call (harness does not re-poison between replays)
  fill_f32<<<grid1((long)N * 4), B, 0, stream>>>(m1, -INFINITY, N * 4);
  fill_f32<<<grid1((long)N * 4), B, 0, stream>>>(s1, 0.0f, N * 4);
  fill_f32<<<grid1((long)N * 32), B, 0, stream>>>(acc1, 0.0f, N * 32);
  fill_f32<<<grid1(N), B, 0, stream>>>(m2, -INFINITY, N);
  fill_f32<<<grid1(N), B, 0, stream>>>(s2, 0.0f, N);
  fill_f32<<<grid1((long)N * 6), B, 0, stream>>>(acc2, 0.0f, N * 6);
  fill_f32<<<grid1((long)G * 6), B, 0, stream>>>(pooled, 0.0f, G * 6);
  fill_f32<<<grid1(G), B, 0, stream>>>(cnt, 0.0f, G);

  // layer 1
  const int tiles  = (N + 15) / 16;
  const int gblocks = (tiles + 7) / 8;            // 8 waves (tiles) per block
  gemm_xfm_l1<<<gblocks, 256, 0, stream>>>(x, Wl1, Wr1, xl1, xr1, N);
  edge_scores<4, 8><<<grid1(Ep), B, 0, stream>>>(xl1, xr1, att1, src, dst, E, N, alpha1, m1);
  edge_exp<4><<<grid1(Ep), B, 0, stream>>>(alpha1, m1, s1, dst, E, N);
  edge_scatter<4, 8><<<grid1(Ep), B, 0, stream>>>(xl1, alpha1, s1, src, dst, E, N, acc1);
  bias_elu32<<<grid1((long)N * 32), B, 0, stream>>>(acc1, b1, N);

  // layer 2
  gemm_xfm_l2<<<gblocks, 256, 0, stream>>>(acc1, Wl2, Wr2, xl2, xr2, N);
  edge_scores<1, 6><<<grid1(Ep), B, 0, stream>>>(xl2, xr2, att2, src, dst, E, N, alpha2, m2);
  edge_exp<1><<<grid1(Ep), B, 0, stream>>>(alpha2, m2, s2, dst, E, N);
  edge_scatter<1, 6><<<grid1(Ep), B, 0, stream>>>(xl2, alpha2, s2, src, dst, E, N, acc2);

  // pooling + log_softmax (b2 folded into mean)
  pool_sum<<<grid1(N), B, 0, stream>>>(acc2, batch, N, pooled, cnt);
  finalize<<<grid1(G), B, 0, stream>>>(pooled, cnt, b2, out, G);
}